// SupernodeFrameTokenizer_65781719106289
// MI455X (gfx1250) — compile-verified
//
#include <hip/hip_runtime.h>
#include <hip/hip_bf16.h>
#include <math.h>

#define BF      8
#define NPTS    16384
#define CH      64
#define M_SN    1024
#define KNN_K   16
#define KGRIP   16
#define KEXTRA  8
#define REMN    (M_SN - KGRIP)   /* 1008 */
#define MASKB   2048             /* mask-id bucket table (ids in [0,32) here) */
#define T1      1024
#define PPT     (NPTS / T1)      /* 16 points per thread in selection kernel */
#define GRIP_R  0.1f
#define FINF    (__builtin_inff())
#define SD_STRIDE 268            /* padded row stride: conflict-free writes+scans */

typedef __attribute__((ext_vector_type(2))) float v2f;
typedef __attribute__((ext_vector_type(8))) float v8f;

#if defined(__has_builtin)
#if __has_builtin(__builtin_amdgcn_wmma_f32_16x16x4_f32)
#define HAVE_WMMA_F32X4 1
#endif
#endif

// ---- output layout (floats, concatenated in reference return order) ----
static constexpr size_t OFF_FEATS = 0;                                   // [8][1024][16][64]
static constexpr size_t OFF_SXYZ  = (size_t)BF * M_SN * KNN_K * CH;      // [8][1024][3]
static constexpr size_t OFF_NMASK = OFF_SXYZ + (size_t)BF * M_SN * 3;    // [8][1024][16]
static constexpr size_t OFF_OIDX  = OFF_NMASK + (size_t)BF * M_SN * KNN_K;
static constexpr size_t OFF_OVAL  = OFF_OIDX + (size_t)BF * M_SN;
static constexpr size_t OFF_OBKT  = OFF_OVAL + (size_t)BF * M_SN;

// Block-wide arg-reduce (min or max, tie-break: lowest index).
__device__ __forceinline__ void argreduce(float* rv, int* ri, int t, float v, int i, bool mx) {
    rv[t] = v; ri[t] = i;
    __syncthreads();
#pragma unroll
    for (int s = T1 / 2; s > 0; s >>= 1) {
        if (t < s) {
            float v2 = rv[t + s]; int i2 = ri[t + s];
            float v1 = rv[t];     int i1 = ri[t];
            bool take = mx ? ((v2 > v1) || (v2 == v1 && i2 < i1))
                           : ((v2 < v1) || (v2 == v1 && i2 < i1));
            if (take) { rv[t] = v2; ri[t] = i2; }
        }
        __syncthreads();
    }
}

// ================= Kernel 1: supernode selection (one block per batch) ==========
__global__ __launch_bounds__(T1) void k_select(
    const float* __restrict__ xyz, const unsigned char* __restrict__ valid,
    const float* __restrict__ state, const int* __restrict__ mask_id,
    const float* __restrict__ extra, const float* __restrict__ glob,
    float* __restrict__ out, float* __restrict__ ws_supxyz)
{
    __shared__ float         s_key[NPTS];        // 64 KB rank keys (CDNA5: 320KB LDS/WGP)
    __shared__ float         s_rv[T1];
    __shared__ int           s_ri[T1];
    __shared__ unsigned int  s_sel[NPTS / 32];   // selected bitmap
    __shared__ unsigned int  s_minpos[MASKB];    // first position per mask id
    __shared__ int           s_maskidx[REMN];
    __shared__ unsigned char s_mkeep[REMN];
    __shared__ int           s_outidx[M_SN];
    __shared__ int           s_first, s_hasr, s_rowv, s_nu;

    const int t = threadIdx.x;
    const int b = blockIdx.x;
    const unsigned char* vrow = valid + (size_t)b * NPTS;

    if (t == 0) { s_first = NPTS; s_hasr = 0; s_rowv = 0; s_nu = 0; }
    for (int i = t; i < NPTS / 32; i += T1) s_sel[i] = 0u;
    for (int i = t; i < MASKB; i += T1) s_minpos[i] = 0xFFFFFFFFu;
    __syncthreads();

    const float gx = state[b * 8 + 0], gy = state[b * 8 + 1], gz = state[b * 8 + 2];

    // gripper distances + row flags
    for (int k = 0; k < PPT; ++k) {
        int p = t + k * T1;
        const float* pp = xyz + ((size_t)b * NPTS + p) * 3;
        float dx = pp[0] - gx, dy = pp[1] - gy, dz = pp[2] - gz;
        float gd = sqrtf(dx * dx + dy * dy + dz * dz);
        s_key[p] = gd;
        if (vrow[p]) {
            atomicMin(&s_first, p);
            atomicOr(&s_rowv, 1);
            if (gd <= GRIP_R) atomicOr(&s_hasr, 1);
        }
    }
    __syncthreads();
    const int  rowv   = s_rowv;
    const int  firstv = (s_first >= NPTS) ? 0 : s_first;
    const int  hasr   = s_hasr;

    // eligibility rank: valid & (in_radius | ~has_radius)
    for (int k = 0; k < PPT; ++k) {
        int p = t + k * T1;
        float gd = s_key[p];
        bool elig = vrow[p] && ((gd <= GRIP_R) || !hasr);
        s_key[p] = elig ? gd : FINF;
    }
    __syncthreads();

    // ---- grip head: 16 iterative arg-mins ----
    for (int j = 0; j < KGRIP; ++j) {
        float lv = FINF; int li = 0;
        for (int k = 0; k < PPT; ++k) {
            int p = t + k * T1;
            float v = s_key[p];
            if (v < lv) { lv = v; li = p; }
        }
        argreduce(s_rv, s_ri, t, lv, li, false);
        if (t == 0) {
            float bv = s_rv[0]; int bi = s_ri[0];
            bool keep = (bv < 1e37f) && rowv;
            int hidx = keep ? bi : firstv;
            if (keep) { s_sel[bi >> 5] |= (1u << (bi & 31)); s_key[bi] = FINF; }
            s_outidx[j] = hidx;
            bool ov = (vrow[hidx] != 0) && rowv;
            out[OFF_OIDX + (size_t)b * M_SN + j] = (float)hidx;
            out[OFF_OVAL + (size_t)b * M_SN + j] = ov ? 1.f : 0.f;
            out[OFF_OBKT + (size_t)b * M_SN + j] = (ov && keep) ? 1.f : -1.f;
        }
        __syncthreads();
    }

    // ---- mask phase: first candidate per mask id, ordered by id ----
    for (int k = 0; k < PPT; ++k) {
        int p = t + k * T1;
        bool sel = (s_sel[p >> 5] >> (p & 31)) & 1u;
        if (vrow[p] && !sel)
            atomicMin(&s_minpos[mask_id[(size_t)b * NPTS + p] & (MASKB - 1)], (unsigned)p);
    }
    __syncthreads();
    if (t == 0) {
        int nu = 0;
        for (int id = 0; id < MASKB && nu < REMN; ++id) {
            unsigned mp = s_minpos[id];
            if (mp != 0xFFFFFFFFu) {
                s_maskidx[nu] = (int)mp;
                s_sel[mp >> 5] |= (1u << (mp & 31));   // selected_after_mask
                ++nu;
            }
        }
        s_nu = nu;
        for (int j2 = 0; j2 < REMN; ++j2) s_mkeep[j2] = (j2 < nu) ? 1 : 0;
    }
    __syncthreads();

    // ---- extra phase: top-8 extra_scores fill slots without mask pick ----
    for (int k = 0; k < PPT; ++k) {
        int p = t + k * T1;
        bool sel = (s_sel[p >> 5] >> (p & 31)) & 1u;
        s_key[p] = (vrow[p] && !sel) ? extra[(size_t)b * NPTS + p] : -FINF;
    }
    __syncthreads();
    for (int j = 0; j < KEXTRA; ++j) {
        float lv = -FINF; int li = 0;
        for (int k = 0; k < PPT; ++k) {
            int p = t + k * T1;
            float v = s_key[p];
            if (v > lv) { lv = v; li = p; }
        }
        argreduce(s_rv, s_ri, t, lv, li, true);
        if (t == 0) {
            float bv = s_rv[0]; int bi = s_ri[0];
            bool ekeep = (bv > -1e37f) && rowv;
            if (bv > -1e37f) s_key[bi] = -FINF;           // consume from top-8 list
            if (!s_mkeep[j] && ekeep) {                   // use_extra
                s_maskidx[j] = bi; s_mkeep[j] = 1;
                s_sel[bi >> 5] |= (1u << (bi & 31));      // selected_for_global
            }
        }
        __syncthreads();
    }

    // ---- global phase: 1008 iterative arg-maxes over global_scores ----
    for (int k = 0; k < PPT; ++k) {
        int p = t + k * T1;
        bool sel = (s_sel[p >> 5] >> (p & 31)) & 1u;
        s_key[p] = (vrow[p] && !sel) ? glob[(size_t)b * NPTS + p] : -FINF;
    }
    __syncthreads();
    for (int j = 0; j < REMN; ++j) {
        float lv = -FINF; int li = 0;
        for (int k = 0; k < PPT; ++k) {
            int p = t + k * T1;
            float v = s_key[p];
            if (v > lv) { lv = v; li = p; }
        }
        argreduce(s_rv, s_ri, t, lv, li, true);
        if (t == 0) {
            float bv = s_rv[0]; int bi = s_ri[0];
            bool gkeep = (bv > -1e37f) && rowv;
            if (bv > -1e37f) s_key[bi] = -FINF;   // list index advances per slot
            int fi, bkt;
            if (s_mkeep[j])      { fi = s_maskidx[j]; bkt = 2; }
            else if (gkeep)      { fi = bi;           bkt = 0; }
            else                 { fi = firstv;       bkt = -1; }
            int m = KGRIP + j;
            s_outidx[m] = fi;
            bool ov = (vrow[fi] != 0) && rowv;
            out[OFF_OIDX + (size_t)b * M_SN + m] = (float)fi;
            out[OFF_OVAL + (size_t)b * M_SN + m] = ov ? 1.f : 0.f;
            out[OFF_OBKT + (size_t)b * M_SN + m] = (float)(ov ? bkt : -1);
        }
        __syncthreads();
    }

    // ---- gather supernode xyz (output + workspace for KNN kernel) ----
    for (int m = t; m < M_SN; m += T1) {
        int idx = s_outidx[m];
        const float* pp = xyz + ((size_t)b * NPTS + idx) * 3;
        float X = pp[0], Y = pp[1], Z = pp[2];
        size_t o = ((size_t)b * M_SN + m) * 3;
        out[OFF_SXYZ + o + 0] = X; out[OFF_SXYZ + o + 1] = Y; out[OFF_SXYZ + o + 2] = Z;
        ws_supxyz[o + 0] = X; ws_supxyz[o + 1] = Y; ws_supxyz[o + 2] = Z;
    }
}

// Async-stage one 256-point xyz chunk (3072 B contiguous, 16B aligned) into LDS.
// Exactly 3 x b64 per thread -> every wave's ASYNCcnt rises by exactly 3.
__device__ __forceinline__ void stage_chunk_async(unsigned ldsbase, const float* gbase, int tid) {
#pragma unroll
    for (int k = 0; k < 3; ++k) {
        unsigned i  = (unsigned)tid + (unsigned)k * 128u;
        unsigned lo = ldsbase + i * 8u;
        unsigned go = i * 8u;
        asm volatile("global_load_async_to_lds_b64 %0, %1, %2"
                     :: "v"(lo), "v"(go), "s"(gbase) : "memory");
    }
}

// ============ Kernel 2: WMMA distance tiles + top-16 KNN + feature gather ========
// Block = 128 threads (4 waves), owns 16 supernodes; each wave computes 16x16
// squared-distance tiles via V_WMMA_F32_16X16X4_F32 (K=3 padded to 4). Point xyz
// is double-buffered in LDS via GLOBAL_LOAD_ASYNC_TO_LDS_B64, pipelined 1 chunk ahead.
__global__ __launch_bounds__(128) void k_knn(
    const float* __restrict__ x, const float* __restrict__ xyz,
    const unsigned char* __restrict__ valid, const float* __restrict__ supxyz,
    float* __restrict__ out)
{
    __shared__ float s_d[16 * SD_STRIDE];        // padded distance staging
    __shared__ float s_xyz[2][256 * 3];          // async double buffer (2 x 3 KB)
    __shared__ float s_ax[16], s_ay[16], s_az[16], s_sa[16];
    __shared__ float s_pv[16 * 128];             // partial top-k merge
    __shared__ int   s_pi[16 * 128];
    __shared__ int   s_nidx[16 * KNN_K];

    const int tid  = threadIdx.x;
    const int b    = blockIdx.y;
    const int m0   = blockIdx.x * 16;
    const int lane = tid & 31;
    const int w    = tid >> 5;

    if (tid < 16) {
        size_t o = ((size_t)b * M_SN + m0 + tid) * 3;
        float X = supxyz[o], Y = supxyz[o + 1], Z = supxyz[o + 2];
        s_ax[tid] = X; s_ay[tid] = Y; s_az[tid] = Z;
        s_sa[tid] = X * X + Y * Y + Z * Z;
    }

    const unsigned ldsb0 = (unsigned)(uintptr_t)(&s_xyz[0][0]);
    const unsigned ldsb1 = (unsigned)(uintptr_t)(&s_xyz[1][0]);
    const float*   gxyz  = xyz + (size_t)b * NPTS * 3;

    // prologue: stage chunk 0 into buffer 0
    stage_chunk_async(ldsb0, gxyz, tid);
    __syncthreads();   // s_a* visible

    const int  n16 = lane & 15;
    const bool hiH = lane >= 16;
    // A operand (16x4 f32): lanes 0-15 hold K=0,1; lanes 16-31 hold K=2,3(=0 pad)
    v2f av;
    av.x = hiH ? s_az[n16] : s_ax[n16];
    av.y = hiH ? 0.f       : s_ay[n16];

    float topv[16]; int topi[16];
#pragma unroll
    for (int k = 0; k < 16; ++k) { topv[k] = FINF; topi[k] = 0; }

    const int row = tid >> 3;   // 8 threads per supernode row in scan phase
    const int sub = tid & 7;

    const int NCH = NPTS / 256;
    for (int ch = 0; ch < NCH; ++ch) {
        const int chb = ch * 256;
        // pipeline: issue next chunk, then wait until only those 3 remain in flight
        if (ch + 1 < NCH) {
            stage_chunk_async((ch & 1) ? ldsb0 : ldsb1, gxyz + (size_t)(chb + 256) * 3, tid);
            asm volatile("s_wait_asynccnt 0x3" ::: "memory");
        } else {
            asm volatile("s_wait_asynccnt 0x0" ::: "memory");
        }
        __syncthreads();   // current chunk staged by all waves
        const float* cx = &s_xyz[ch & 1][0];

#pragma unroll
        for (int tile = 0; tile < 4; ++tile) {
            int pcol = w * 64 + tile * 16 + n16;
            float px = cx[pcol * 3 + 0];
            float py = cx[pcol * 3 + 1];
            float pz = cx[pcol * 3 + 2];
            bool  pv = valid[(size_t)b * NPTS + chb + pcol] != 0;
            // B operand (4x16 f32): lanes 0-15 rows K=0,1; lanes 16-31 rows K=2,3(=0)
            v2f bv;
            bv.x = hiH ? pz  : px;
            bv.y = hiH ? 0.f : py;
            float carr[8];
#ifdef HAVE_WMMA_F32X4
            v8f c = {};
            c = __builtin_amdgcn_wmma_f32_16x16x4_f32(
                    false, av, false, bv, (short)0, c, false, false);
#pragma unroll
            for (int r = 0; r < 8; ++r) carr[r] = c[r];
#else
#pragma unroll
            for (int r = 0; r < 8; ++r) {
                int Mr = r + (hiH ? 8 : 0);
                carr[r] = s_ax[Mr] * px + s_ay[Mr] * py + s_az[Mr] * pz;
            }
#endif
            float sb = px * px + py * py + pz * pz;
#pragma unroll
            for (int r = 0; r < 8; ++r) {
                int Mr = r + (hiH ? 8 : 0);
                // rank by clamped squared distance (monotonic in distance; sqrt elided)
                float d2 = fmaxf(s_sa[Mr] + sb - 2.0f * carr[r], 0.f);
                s_d[Mr * SD_STRIDE + pcol] = pv ? d2 : FINF;   // branchless select
            }
        }
        __syncthreads();
        // per-thread running top-16; col = sub + cc*8 -> 32 distinct banks per wave
        for (int cc = 0; cc < 32; ++cc) {
            int col = sub + cc * 8;
            float d = s_d[row * SD_STRIDE + col];
            if (d < topv[15]) {
                float cv = d; int ci = chb + col;
#pragma unroll
                for (int q = 0; q < 16; ++q) {   // unrolled sorted insert (regs only)
                    float tv = topv[q]; int ti = topi[q];
                    bool sw = cv < tv;
                    topv[q] = sw ? cv : tv; topi[q] = sw ? ci : ti;
                    cv = sw ? tv : cv;      ci = sw ? ti : ci;
                }
            }
        }
        __syncthreads();   // also fences s_xyz buffer reuse
    }

    // ---- merge 8 partial lists per row ----
    int pbase = (row * 8 + sub) * 16;
#pragma unroll
    for (int k = 0; k < 16; ++k) { s_pv[pbase + k] = topv[k]; s_pi[pbase + k] = topi[k]; }
    __syncthreads();
    if (tid < 16) {
        int r = tid;
        for (int k = 0; k < KNN_K; ++k) {
            float best = FINF; int bi = 0x7FFFFFFF; int bs = 0;
            for (int e = 0; e < 128; ++e) {
                float v = s_pv[r * 128 + e]; int ii = s_pi[r * 128 + e];
                if (v < best || (v == best && ii < bi)) { best = v; bi = ii; bs = e; }
            }
            s_pv[r * 128 + bs] = FINF; s_pi[r * 128 + bs] = 0x7FFFFFFF;
            bool fin = best < 1e37f;
            s_nidx[r * KNN_K + k] = (bi == 0x7FFFFFFF) ? 0 : bi;
            out[OFF_NMASK + ((size_t)b * M_SN + m0 + r) * KNN_K + k] = fin ? 1.f : 0.f;
        }
    }
    __syncthreads();

    // ---- gather neighbor features (float4 streams) ----
    for (int pair = tid; pair < 16 * KNN_K; pair += 128) {
        int r = pair >> 4, k = pair & 15;
        int nidx = s_nidx[pair];
        const float* src = x + ((size_t)b * NPTS + nidx) * CH;
        __builtin_prefetch(src, 0, 1);
        float* dst = out + OFF_FEATS + (((size_t)b * M_SN + m0 + r) * KNN_K + k) * CH;
        const float4* s4 = (const float4*)src;
        float4*       d4 = (float4*)dst;
#pragma unroll
        for (int q = 0; q < CH / 4; ++q) d4[q] = s4[q];
    }
}

extern "C" void kernel_launch(void* const* d_in, const int* in_sizes, int n_in,
                              void* d_out, int out_size, void* d_ws, size_t ws_size,
                              hipStream_t stream) {
    const float*         x       = (const float*)d_in[0];
    const float*         xyz     = (const float*)d_in[1];
    const unsigned char* valid   = (const unsigned char*)d_in[2];   // jnp bool -> 1 byte
    const float*         state   = (const float*)d_in[3];
    const int*           mask_id = (const int*)d_in[4];
    const float*         extra   = (const float*)d_in[5];
    const float*         glob    = (const float*)d_in[6];
    float*               out     = (float*)d_out;
    float*               supxyz  = (float*)d_ws;   // 8*1024*3 floats = 96 KB scratch

    k_select<<<BF, T1, 0, stream>>>(xyz, valid, state, mask_id, extra, glob, out, supxyz);
    dim3 g2(M_SN / 16, BF);
    k_knn<<<g2, 128, 0, stream>>>(x, xyz, valid, supxyz, out);
}